// ActiveBoundaryLoss_14010183320289
// MI455X (gfx1250) — compile-verified
//
#include <hip/hip_runtime.h>

// ---------------- problem constants ----------------
#define NCLS 19
#define Bn   4
#define Hh   512
#define Ww   512
#define HW   (Hh * Ww)            // 262144
#define NPIX (Bn * HW)            // 1048576
#define BLK  256
#define GRD  (NPIX / BLK)         // 4096 (multiple of 64 -> guard-free WMMA reduce)
#define NBINS 129
#define MAXDIS 100000.0f
#define MAXCLIP 20.0f
#define LBPOS 0.8f
#define LBNEG 0.025f
#define BORDER_F 0.1f

typedef __attribute__((ext_vector_type(2))) float v2f;
typedef __attribute__((ext_vector_type(8))) float v8f;

// neighbor order must match the reference _DX/_DY lists
// _DX indexes rows (H), _DY indexes cols (W); index 8 = self
__device__ __constant__ int c_dx[9] = {1, -1, 0, 0, -1, 1, -1, 1, 0};
__device__ __constant__ int c_dy[9] = {0, 0, -1, 1, 1, 1, -1, -1, 0};

// ---------------- pass 1: lse, S = sum(p*l), target CE ----------------
__global__ void k_pass1(const float* __restrict__ x, const int* __restrict__ tgt,
                        float* __restrict__ lse, float* __restrict__ Sarr,
                        float* __restrict__ ce_part) {
  const int idx = blockIdx.x * BLK + threadIdx.x;
  const int b = idx / HW;
  const int p = idx - b * HW;
  const float* xb = x + (size_t)b * NCLS * HW + p;

  float v[NCLS];
  float m = -3.4e38f;
#pragma unroll
  for (int c = 0; c < NCLS; ++c) { v[c] = xb[(size_t)c * HW]; m = fmaxf(m, v[c]); }
  float se = 0.f;
#pragma unroll
  for (int c = 0; c < NCLS; ++c) se += __expf(v[c] - m);
  const float L = m + __logf(se);
  float Sv = 0.f;
#pragma unroll
  for (int c = 0; c < NCLS; ++c) { float l = v[c] - L; Sv += __expf(l) * l; }
  lse[idx] = L;
  Sarr[idx] = Sv;

  // target CE: -(x[clip(t)] - lse), zero where t == 255 (ignore)
  const int t = tgt[idx];
  int tc = t < 0 ? 0 : (t > NCLS - 1 ? NCLS - 1 : t);
  float xs = v[0];
#pragma unroll
  for (int c = 1; c < NCLS; ++c) xs = (c == tc) ? v[c] : xs;
  float ce = (t == 255) ? 0.f : (L - xs);

  __shared__ float sd[BLK];
  sd[threadIdx.x] = ce;
  __syncthreads();
  for (int s = BLK / 2; s > 0; s >>= 1) {
    if (threadIdx.x < s) sd[threadIdx.x] += sd[threadIdx.x + s];
    __syncthreads();
  }
  if (threadIdx.x == 0) ce_part[blockIdx.x] = sd[0];
}

// ---------------- pass 2: klc = kl_ud + kl_lr, histogram for eps ----------------
__global__ void k_pass2(const float* __restrict__ x, const float* __restrict__ lse,
                        const float* __restrict__ Sarr, float* __restrict__ klc,
                        int* __restrict__ hist) {
  const int idx = blockIdx.x * BLK + threadIdx.x;
  const int b = idx / HW;
  const int p = idx - b * HW;
  const int y = p / Ww;
  const int xc = p - y * Ww;
  const float* xa = x + (size_t)b * NCLS * HW + p;

  const float La = lse[idx];
  const float Sa = Sarr[idx];
  float pa[NCLS];
#pragma unroll
  for (int c = 0; c < NCLS; ++c) pa[c] = __expf(xa[(size_t)c * HW] - La);

  float kl = 0.f;
  if (y < Hh - 1) {                 // KL(self || down): S_a - <p_a, l_down>
    const float* xd = xa + Ww;
    const float Ld = lse[idx + Ww];
    float dot = 0.f;
#pragma unroll
    for (int c = 0; c < NCLS; ++c) dot += pa[c] * (xd[(size_t)c * HW] - Ld);
    kl += Sa - dot;
  }
  if (xc < Ww - 1) {                // KL(self || right)
    const float* xr = xa + 1;
    const float Lr = lse[idx + 1];
    float dot = 0.f;
#pragma unroll
    for (int c = 0; c < NCLS; ++c) dot += pa[c] * (xr[(size_t)c * HW] - Lr);
    kl += Sa - dot;
  }
  klc[idx] = kl;

  // bin = number of thresholds eps_k = 1e-5f * 1.2f^k this value exceeds
  int cnt = 0;
  float e = 1e-5f;
  while (cnt < NBINS - 1 && kl > e) { ++cnt; e *= 1.2f; }
  atomicAdd(&hist[cnt], 1);
}

// ---------------- eps search: exact replica of the reference while-loop ----------------
__global__ void k_eps(const int* __restrict__ hist, float* __restrict__ epsout) {
  if (threadIdx.x == 0 && blockIdx.x == 0) {
    const float maxN = (float)(Hh * Ww) * 0.01f;   // 2621.44
    int cnt = 0;                                    // cnt_0 = #elements with klc > eps_0
    for (int bn = 1; bn < NBINS; ++bn) cnt += hist[bn];
    float e = 1e-5f;
    for (int k = 0; k < NBINS - 1; ++k) {
      if (!((float)cnt > maxN)) break;
      e *= 1.2f;
      cnt -= hist[k + 1];                           // cnt_{k+1} = cnt_k - hist[k+1]
    }
    epsout[0] = e;
  }
}

// ---------------- pass 4: dilation, argmin dist, kl8 + smoothed CE ----------------
__global__ void k_pass4(const float* __restrict__ x, const float* __restrict__ dist,
                        const float* __restrict__ lse, const float* __restrict__ Sarr,
                        const float* __restrict__ klc, const float* __restrict__ epsp,
                        float* __restrict__ lsce_part, float* __restrict__ w_part,
                        float* __restrict__ cnt_part) {
  const int idx = blockIdx.x * BLK + threadIdx.x;
  const int b = idx / HW;
  const int p = idx - b * HW;
  const int y = p / Ww;
  const int xc = p - y * Ww;
  const float eps = epsp[0];
  const float* klcb = klc + (size_t)b * HW;
  const float* xa = x + (size_t)b * NCLS * HW + p;

  __builtin_prefetch(xa, 0, 3);   // global_prefetch_b8: warm self logits for valid path

  // pb = 3x3 dilation of (klc > eps), zero-padded (SAME conv with ones)
  bool pb = false;
#pragma unroll
  for (int dyy = -1; dyy <= 1; ++dyy)
#pragma unroll
    for (int dxx = -1; dxx <= 1; ++dxx) {
      int yy = y + dyy, xx = xc + dxx;
      if (yy >= 0 && yy < Hh && xx >= 0 && xx < Ww) pb |= (klcb[yy * Ww + xx] > eps);
    }

  float lsce_c = 0.f, w_c = 0.f;
  const float pbf = pb ? 1.f : 0.f;

  if (pb) {
    const float* db = dist + (size_t)b * HW;
    // argmin over 9 shifted dists (OOB -> MAXDIS), first-min wins
    float best = 3.4e38f;
    int dgt = 0;
#pragma unroll
    for (int k = 0; k < 9; ++k) {
      int yy = y + c_dx[k], xx = xc + c_dy[k];
      float d = (yy >= 0 && yy < Hh && xx >= 0 && xx < Ww) ? db[yy * Ww + xx] : MAXDIS;
      if (d < best) { best = d; dgt = k; }
    }
    if (dgt != 8) {                 // valid = pb & (dgt != 8)
      const float La = lse[idx];
      float la[NCLS];
#pragma unroll
      for (int c = 0; c < NCLS; ++c) la[c] = xa[(size_t)c * HW] - La;

      float kl8[8];
#pragma unroll
      for (int j = 0; j < 8; ++j) {   // KL(neighbor || self), edge-clamped neighbor
        int yy = y + c_dx[j]; yy = yy < 0 ? 0 : (yy > Hh - 1 ? Hh - 1 : yy);
        int xx = xc + c_dy[j]; xx = xx < 0 ? 0 : (xx > Ww - 1 ? Ww - 1 : xx);
        const int nidx = b * HW + yy * Ww + xx;
        const float* xn = x + (size_t)b * NCLS * HW + yy * Ww + xx;
        const float Ln = lse[nidx];
        float dot = 0.f;
#pragma unroll
        for (int c = 0; c < NCLS; ++c) dot += __expf(xn[(size_t)c * HW] - Ln) * la[c];
        kl8[j] = Sarr[nidx] - dot;
      }
      // log_softmax over 8, label-smoothed CE
      float m8 = kl8[0];
#pragma unroll
      for (int j = 1; j < 8; ++j) m8 = fmaxf(m8, kl8[j]);
      float se8 = 0.f, sumkl = 0.f;
#pragma unroll
      for (int j = 0; j < 8; ++j) { se8 += __expf(kl8[j] - m8); sumkl += kl8[j]; }
      const float lse8 = m8 + __logf(se8);
      float klsel = kl8[0];
#pragma unroll
      for (int j = 1; j < 8; ++j) klsel = (j == dgt) ? kl8[j] : klsel;
      const float logqd = klsel - lse8;
      const float sumlogq = sumkl - 8.f * lse8;
      lsce_c = -(LBPOS - LBNEG) * logqd - LBNEG * sumlogq;
      w_c = fminf(db[y * Ww + xc], MAXCLIP) / MAXCLIP;
    }
  }

  __shared__ float s1[BLK], s2[BLK], s3[BLK];
  s1[threadIdx.x] = lsce_c;
  s2[threadIdx.x] = w_c;
  s3[threadIdx.x] = pbf;
  __syncthreads();
  for (int s = BLK / 2; s > 0; s >>= 1) {
    if (threadIdx.x < s) {
      s1[threadIdx.x] += s1[threadIdx.x + s];
      s2[threadIdx.x] += s2[threadIdx.x + s];
      s3[threadIdx.x] += s3[threadIdx.x + s];
    }
    __syncthreads();
  }
  if (threadIdx.x == 0) {
    lsce_part[blockIdx.x] = s1[0];
    w_part[blockIdx.x]    = s2[0];
    cnt_part[blockIdx.x]  = s3[0];
  }
}

// ---------------- final: WMMA-chained f32 reduction + scalar combine ----------------
// V_WMMA_F32_16X16X4_F32 with B = ones: D[i][j] = sum_k A[i][k] + C[i][j].
// Total over all D rows == sum of ALL 64 A-slot values, independent of which
// slot each value occupies -> each lane can load a CONTIGUOUS float2 (b64,
// fully coalesced, no guards since n % 64 == 0).
// C/D (16x16 f32): lane j<16 holds rows 0-7 of column j; lane j>=16 rows 8-15,
// so lane0's 8 accs + lane16's 8 accs cover every row exactly once.
__device__ inline float wave_wmma_sum(const float* __restrict__ arr, int n, int lane) {
  v8f c = {0.f, 0.f, 0.f, 0.f, 0.f, 0.f, 0.f, 0.f};
  v2f ones; ones.x = 1.f; ones.y = 1.f;
  const v2f* __restrict__ a2 = (const v2f*)arr;
  const int npairs = n >> 1;              // n is a multiple of 64 -> npairs % 32 == 0
#pragma unroll 4
  for (int base = lane; base < npairs; base += 32) {
    v2f a = a2[base];                     // global_load_b64, coalesced
    c = __builtin_amdgcn_wmma_f32_16x16x4_f32(false, a, false, ones,
                                              (short)0, c, false, false);
  }
  float s = c[0] + c[1] + c[2] + c[3] + c[4] + c[5] + c[6] + c[7];
  return __shfl(s, 0, 32) + __shfl(s, 16, 32);
}

__global__ void k_final(const float* __restrict__ ce_part,
                        const float* __restrict__ lsce_part,
                        const float* __restrict__ w_part,
                        const float* __restrict__ cnt_part,
                        int g, float* __restrict__ out) {
  const int lane = threadIdx.x;            // exactly one wave32, EXEC all ones
  const float ce  = wave_wmma_sum(ce_part, g, lane);
  const float l1  = wave_wmma_sum(lsce_part, g, lane);
  const float l2  = wave_wmma_sum(w_part, g, lane);
  const float cnt = wave_wmma_sum(cnt_part, g, lane);
  if (lane == 0) {
    const float border = (cnt > 1.f) ? (l1 * l2) : 0.f;
    out[0] = BORDER_F * border + ce;
  }
}

// ---------------- launch ----------------
extern "C" void kernel_launch(void* const* d_in, const int* in_sizes, int n_in,
                              void* d_out, int out_size, void* d_ws, size_t ws_size,
                              hipStream_t stream) {
  (void)in_sizes; (void)n_in; (void)out_size; (void)ws_size;
  const float* slices = (const float*)d_in[0];   // (4,19,512,512) f32
  const float* dist   = (const float*)d_in[1];   // (4,1,512,512)  f32
  const int*   tgt    = (const int*)d_in[2];     // (4,1,512,512)  i32
  float* out = (float*)d_out;

  float* ws      = (float*)d_ws;
  float* lse     = ws;                         // NPIX
  float* Sarr    = ws + (size_t)NPIX;          // NPIX
  float* klc     = ws + (size_t)2 * NPIX;      // NPIX
  float* ce_part   = ws + (size_t)3 * NPIX;    // GRD
  float* lsce_part = ce_part + GRD;            // GRD
  float* w_part    = lsce_part + GRD;          // GRD
  float* cnt_part  = w_part + GRD;             // GRD
  float* epsb      = cnt_part + GRD;           // 1
  int*   hist      = (int*)(epsb + 1);         // NBINS

  hipMemsetAsync(hist, 0, NBINS * sizeof(int), stream);
  k_pass1<<<GRD, BLK, 0, stream>>>(slices, tgt, lse, Sarr, ce_part);
  k_pass2<<<GRD, BLK, 0, stream>>>(slices, lse, Sarr, klc, hist);
  k_eps<<<1, 32, 0, stream>>>(hist, epsb);
  k_pass4<<<GRD, BLK, 0, stream>>>(slices, dist, lse, Sarr, klc, epsb,
                                   lsce_part, w_part, cnt_part);
  k_final<<<1, 32, 0, stream>>>(ce_part, lsce_part, w_part, cnt_part, GRD, out);
}